// PinUtilization_12575664242758
// MI455X (gfx1250) — compile-verified
//
#include <hip/hip_runtime.h>
#include <stdint.h>

// Problem constants (from the reference)
#define NBX 1024
#define NBY 1024
// bin size = (1024-0)/1024 = 1.0 exactly -> /bsize, *bsize fold away.
// Output scale = 1/(bsx*bsy*UNIT_PIN_CAPACITY) = 10, folded into density.
#define OUT_SCALE 10.0f
#define STRETCH 1.4142135623730951f
#define TPB 256
#define GRID_BLOCKS 1024

// ---------------------------------------------------------------------------
// Zero the 1024x1024 f32 output (harness poisons d_out once; re-zero every
// launch so graph replays stay deterministic).
// ---------------------------------------------------------------------------
__global__ __launch_bounds__(TPB) void zero_out_kernel(float4* __restrict__ out) {
  out[blockIdx.x * TPB + threadIdx.x] = make_float4(0.f, 0.f, 0.f, 0.f);
}

// ---------------------------------------------------------------------------
// CDNA5 async global->LDS staging (ASYNCcnt path, CDNA5 ISA
// §4). GVS mode: mem_addr = SGPR_u64 + VGPR_i32; per-lane LDS byte address in
// the vdst VGPR. th:TH_LOAD_NT keeps the 40MB one-shot input stream from
// evicting the 4MB output grid we want hot in the 192MB L2 for atomics.
// ---------------------------------------------------------------------------
__device__ __forceinline__ void async_load_f32(uint32_t lds_byte_addr,
                                               const float* __restrict__ base,
                                               uint32_t byte_off) {
  asm volatile("global_load_async_to_lds_b32 %0, %1, %2 th:TH_LOAD_NT"
               :
               : "v"(lds_byte_addr), "v"(byte_off), "s"(base)
               : "memory");
}

__device__ __forceinline__ void wait_async_0() {
  asm volatile("s_wait_asynccnt 0" ::: "memory");
}

// After this, the 5 staged values are architecturally in VGPRs; the memory
// clobber also stops the compiler from re-reading LDS after the next issue.
__device__ __forceinline__ void wait_ds_0() {
  asm volatile("s_wait_dscnt 0" ::: "memory");
}

// ---------------------------------------------------------------------------
// Main scatter kernel. One node per thread per tile; grid-stride over nodes.
// Pipeline per iteration:
//   wait_async(0)  -> tile i staged in LDS
//   ds_load x5     -> values to VGPRs; wait_ds(0)
//   issue tile i+1 -> streams from HBM while we compute/atomic tile i
//   compute + up to 16 (typically ~7) global_atomic_add_f32 into L2
// ~13M f32 L2 atomics dominate; input streaming (40MB, 1.7us at 23.3TB/s) is
// fully hidden behind them by the async prefetch.
// ---------------------------------------------------------------------------
__global__ __launch_bounds__(TPB) void pin_util_kernel(
    const float* __restrict__ g_x,   // pos[0 .. N)
    const float* __restrict__ g_y,   // pos[N .. 2N)
    const float* __restrict__ g_nsx,
    const float* __restrict__ g_nsy,
    const float* __restrict__ g_w,
    float* __restrict__ out,
    int n) {
  __shared__ float sb[5][TPB];  // single buffer: reads drain before re-issue

  const int tid = threadIdx.x;
  const int stride = (int)(gridDim.x * blockDim.x);
  int idx = (int)(blockIdx.x * blockDim.x) + tid;
  if (idx >= n) return;

  auto issue_tile = [&](int i) {
    uint32_t off = (uint32_t)i << 2;  // byte offset into each f32 stream
    async_load_f32((uint32_t)(uintptr_t)&sb[0][tid], g_x, off);
    async_load_f32((uint32_t)(uintptr_t)&sb[1][tid], g_y, off);
    async_load_f32((uint32_t)(uintptr_t)&sb[2][tid], g_nsx, off);
    async_load_f32((uint32_t)(uintptr_t)&sb[3][tid], g_nsy, off);
    async_load_f32((uint32_t)(uintptr_t)&sb[4][tid], g_w, off);
  };

  issue_tile(idx);  // prologue

  for (; idx < n; idx += stride) {
    wait_async_0();  // tile i landed in LDS

    float x   = sb[0][tid];
    float y   = sb[1][tid];
    float nsx = sb[2][tid];
    float nsy = sb[3][tid];
    float w   = sb[4][tid];
    wait_ds_0();  // values in VGPRs -> safe to overwrite the LDS buffer

    // Prefetch tile i+1 (clamped tail index: harmless dup, never consumed).
    int nidx = idx + stride;
    issue_tile(nidx < n ? nidx : (n - 1));

    // --- compute (bin size == 1.0) ---
    float hx = 0.5f * fmaxf(nsx, STRETCH);
    float hy = 0.5f * fmaxf(nsy, STRETCH);
    float cx = x + 0.5f * nsx;
    float cy = y + 0.5f * nsy;
    float xmin = cx - hx, xmax = cx + hx;
    float ymin = cy - hy, ymax = cy + hy;

    int blx = (int)floorf(xmin); blx = blx > 0 ? blx : 0;
    int bhx = (int)floorf(xmax) + 1; bhx = bhx < NBX ? bhx : NBX;
    int bly = (int)floorf(ymin); bly = bly > 0 ? bly : 0;
    int bhy = (int)floorf(ymax) + 1; bhy = bhy < NBY ? bhy : NBY;

    float dens = (w * OUT_SCALE) / (4.0f * hx * hy);

    float oxv[4], oyv[4];
#pragma unroll
    for (int i = 0; i < 4; ++i) {
      int b = blx + i;
      float ov = fminf((float)(b + 1), xmax) - fmaxf((float)b, xmin);
      oxv[i] = (b < bhx) ? ov : 0.0f;
      b = bly + i;
      ov = fminf((float)(b + 1), ymax) - fmaxf((float)b, ymin);
      oyv[i] = (b < bhy) ? ov : 0.0f;
    }

    // Scatter: stretched width <= 2 bins -> typically ~2-3 valid bins/axis;
    // skipping zero overlaps cuts atomics ~2.4x vs blind 4x4.
#pragma unroll
    for (int i = 0; i < 4; ++i) {
      if (oxv[i] == 0.0f) continue;
      float fx = dens * oxv[i];
      float* row = out + (((blx + i) << 10) + bly);
#pragma unroll
      for (int j = 0; j < 4; ++j) {
        if (oyv[j] == 0.0f) continue;
        atomicAdd(row + j, fx * oyv[j]);  // global_atomic_add_f32, L2-resident
      }
    }
  }
  // Trailing prefetch drains at s_endpgm's implicit wait-idle.
}

// ---------------------------------------------------------------------------
extern "C" void kernel_launch(void* const* d_in, const int* in_sizes, int n_in,
                              void* d_out, int out_size, void* d_ws, size_t ws_size,
                              hipStream_t stream) {
  const float* pos = (const float*)d_in[0];   // 2*NUM_NODES
  const float* nsx = (const float*)d_in[1];   // NUM_NODES
  const float* nsy = (const float*)d_in[2];   // NUM_NODES
  const float* w   = (const float*)d_in[3];   // NUM_PHYSICAL
  float* out = (float*)d_out;

  const int num_nodes = in_sizes[0] / 2;
  const int n_phys = in_sizes[3];

  zero_out_kernel<<<(NBX * NBY) / 4 / TPB, TPB, 0, stream>>>((float4*)out);
  pin_util_kernel<<<GRID_BLOCKS, TPB, 0, stream>>>(
      pos, pos + num_nodes, nsx, nsy, w, out, n_phys);
}